// GNNEncoder_52561809768660
// MI455X (gfx1250) — compile-verified
//
#include <hip/hip_runtime.h>
#include <hip/hip_bf16.h>

typedef __attribute__((ext_vector_type(2))) float v2f;
typedef __attribute__((ext_vector_type(8))) float v8f;

#define D_FEAT 128

// ---------------------------------------------------------------------------
// Zero a float buffer (n4 = count of float4)
// ---------------------------------------------------------------------------
__global__ void zero_kernel(float* __restrict__ p, long n4) {
    long i = (long)blockIdx.x * blockDim.x + threadIdx.x;
    long stride = (long)gridDim.x * blockDim.x;
    float4 z = make_float4(0.f, 0.f, 0.f, 0.f);
    for (; i < n4; i += stride) ((float4*)p)[i] = z;
}

// ---------------------------------------------------------------------------
// In-degree counts: one thread per edge
// ---------------------------------------------------------------------------
__global__ void count_kernel(const int* __restrict__ dst, float* __restrict__ cnt, int E) {
    int e = blockIdx.x * blockDim.x + threadIdx.x;
    if (e < E) atomicAdd(cnt + dst[e], 1.0f);
}

// cnt[i] -> 1 / max(cnt[i], 1)  (in place)
__global__ void inv_kernel(float* __restrict__ cnt, int N) {
    int i = blockIdx.x * blockDim.x + threadIdx.x;
    if (i < N) cnt[i] = 1.0f / fmaxf(cnt[i], 1.0f);
}

// ---------------------------------------------------------------------------
// Edge scatter-add: one wave (32 lanes) per edge; lane handles 4 contiguous
// floats => the 512B feature row is fully coalesced. agg[dst] += feat[src].
// All feature tables (~51MB each) are resident in the 192MB L2, so both the
// gather and the f32 atomics resolve in L2.
// ---------------------------------------------------------------------------
__global__ void scatter_kernel(const float* __restrict__ feat,
                               const int* __restrict__ src,
                               const int* __restrict__ dst,
                               float* __restrict__ agg, int E) {
    int gtid = blockIdx.x * blockDim.x + threadIdx.x;
    int e    = gtid >> 5;          // wave index == edge index
    int lane = gtid & 31;
    if (e >= E) return;
    int s = src[e];                // scalar broadcast load
    int d = dst[e];
    const float4 v = *(const float4*)(feat + (size_t)s * D_FEAT + lane * 4);
    float* o = agg + (size_t)d * D_FEAT + lane * 4;
    atomicAdd(o + 0, v.x);         // global_atomic_add_f32
    atomicAdd(o + 1, v.y);
    atomicAdd(o + 2, v.z);
    atomicAdd(o + 3, v.w);
}

// ---------------------------------------------------------------------------
// Fused SAGE layer GEMM via V_WMMA_F32_16X16X4_F32:
//   out[16 rows x 128 cols per block] =
//       (agg*inv_cnt) @ Wl^T + feat @ Wr^T + b   (+ optional relu)
// Block = 256 threads = 8 waves; wave w computes the 16x16 tile at col w*16.
// A-tiles staged in LDS (coalesced); B fragments from global (Wl/Wr = 64KB
// each -> WGP$/L2 resident). Two independent accumulators break the WMMA
// RAW chain so consecutive WMMAs can pipeline in the XDL unit.
//
// Fragment layouts per CDNA5 ISA 7.12.2:
//   A 16x4 f32 (2 VGPRs): m = lane&15, k = k0 + 2*(lane>>4) + {0,1}
//   B 4x16 f32 (2 VGPRs): n = lane&15, k = k0 + 2*(lane>>4) + {0,1}
//   C/D 16x16 f32 (8 VGPRs): n = lane&15, m = i + 8*(lane>>4)
// ---------------------------------------------------------------------------
__global__ __launch_bounds__(256) void sage_gemm_kernel(
    const float* __restrict__ agg, const float* __restrict__ feat,
    const float* __restrict__ Wl, const float* __restrict__ Wr,
    const float* __restrict__ bias, const float* __restrict__ inv_cnt,
    float* __restrict__ out, int do_relu) {

    __shared__ float sA[16 * D_FEAT];   // scaled aggregate tile
    __shared__ float sX[16 * D_FEAT];   // self-feature tile

    const int row0 = blockIdx.x * 16;   // N = 100000 is a multiple of 16
    const int tid  = threadIdx.x;

    // Coalesced stage-in of the 16x128 A-tiles; fold the 1/deg scaling here.
#pragma unroll
    for (int i = 0; i < 8; ++i) {
        int elem = tid + i * 256;
        int r = elem >> 7;              // 0..15
        int c = elem & 127;
        size_t g = (size_t)(row0 + r) * D_FEAT + c;
        sA[elem] = agg[g] * inv_cnt[row0 + r];
        sX[elem] = feat[g];
    }
    __syncthreads();

    const int wave = tid >> 5;
    const int lane = tid & 31;
    const int col0 = wave * 16;
    const int half = lane >> 4;         // K-pair selector
    const int idx  = lane & 15;         // M for A-frags, N for B-frags

    const float bval = bias[col0 + idx];

    v8f accL = {};                      // agg @ Wl^T chain
    v8f accR = {};                      // feat @ Wr^T chain (independent)
#pragma unroll
    for (int k0 = 0; k0 < D_FEAT; k0 += 4) {
        const int kk = k0 + 2 * half;
        v2f aA = *(const v2f*)(sA + idx * D_FEAT + kk);                  // ds_load_b64
        v2f aX = *(const v2f*)(sX + idx * D_FEAT + kk);
        v2f bL = *(const v2f*)(Wl + (size_t)(col0 + idx) * D_FEAT + kk); // B[k][n]=Wl[n][k]
        v2f bR = *(const v2f*)(Wr + (size_t)(col0 + idx) * D_FEAT + kk);
        accL = __builtin_amdgcn_wmma_f32_16x16x4_f32(false, aA, false, bL,
                                                     (short)0, accL, false, false);
        accR = __builtin_amdgcn_wmma_f32_16x16x4_f32(false, aX, false, bR,
                                                     (short)0, accR, false, false);
    }

    // Epilogue: combine chains, bias (+relu), write 16x16 tile
#pragma unroll
    for (int i = 0; i < 8; ++i) {
        int m = i + 8 * half;
        float v = accL[i] + accR[i] + bval;
        if (do_relu) v = fmaxf(v, 0.0f);
        out[(size_t)(row0 + m) * D_FEAT + col0 + idx] = v;
    }
}

// ---------------------------------------------------------------------------
// Launch: inputs per setup_inputs() order:
//   0:x [N,128] f32  1:edge_index [2,E] int  2:Wl1  3:Wr1  4:b1  5:Wl2 6:Wr2 7:b2
// ---------------------------------------------------------------------------
extern "C" void kernel_launch(void* const* d_in, const int* in_sizes, int n_in,
                              void* d_out, int out_size, void* d_ws, size_t ws_size,
                              hipStream_t stream) {
    const float* x    = (const float*)d_in[0];
    const int*   ei   = (const int*)d_in[1];
    const float* Wl1  = (const float*)d_in[2];
    const float* Wr1  = (const float*)d_in[3];
    const float* b1   = (const float*)d_in[4];
    const float* Wl2  = (const float*)d_in[5];
    const float* Wr2  = (const float*)d_in[6];
    const float* b2   = (const float*)d_in[7];
    float*       out  = (float*)d_out;

    const int N = in_sizes[0] / D_FEAT;     // 100000
    const int E = in_sizes[1] / 2;          // 1600000
    const int* src = ei;
    const int* dst = ei + E;

    // Workspace layout (floats): agg [N*128] | h [N*128] | cnt [N]
    float* agg = (float*)d_ws;
    float* h   = agg + (size_t)N * D_FEAT;
    float* cnt = h + (size_t)N * D_FEAT;

    const int TB = 256;
    const long aggN4 = (long)N * D_FEAT / 4;
    const int zeroBlocks = 2048;

    // ---- degrees + layer 1 aggregation ----
    zero_kernel<<<zeroBlocks, TB, 0, stream>>>(agg, aggN4);
    zero_kernel<<<(N / 4 + TB - 1) / TB, TB, 0, stream>>>(cnt, N / 4);
    count_kernel<<<(E + TB - 1) / TB, TB, 0, stream>>>(dst, cnt, E);
    scatter_kernel<<<((long)E * 32 + TB - 1) / TB, TB, 0, stream>>>(x, src, dst, agg, E);
    inv_kernel<<<(N + TB - 1) / TB, TB, 0, stream>>>(cnt, N);

    // ---- layer 1: h = relu(aggS @ Wl1^T + x @ Wr1^T + b1) ----
    sage_gemm_kernel<<<N / 16, TB, 0, stream>>>(agg, x, Wl1, Wr1, b1, cnt, h, 1);

    // ---- layer 2 aggregation + GEMM ----
    zero_kernel<<<zeroBlocks, TB, 0, stream>>>(agg, aggN4);
    scatter_kernel<<<((long)E * 32 + TB - 1) / TB, TB, 0, stream>>>(h, src, dst, agg, E);
    sage_gemm_kernel<<<N / 16, TB, 0, stream>>>(agg, h, Wl2, Wr2, b2, cnt, out, 0);
}